// GAT_Net_65326452572654
// MI455X (gfx1250) — compile-verified
//
#include <hip/hip_runtime.h>
#include <hip/hip_bf16.h>
#include <math.h>

typedef __attribute__((ext_vector_type(2))) float v2f;
typedef __attribute__((ext_vector_type(8))) float v8f;

#define F_IN   165
#define KPAD   168      // F_IN rounded up to multiple of 4
#define LDS_STRIDE 172  // padded row stride (gcd(172,64)=4 -> 16 distinct banks)
#define HC     256      // heads * out_ch = 4*64
#define NEG_SLOPE 0.2f

// ---------------- helpers ----------------

__device__ __forceinline__ float leaky(float a) {
    return a > 0.0f ? a : NEG_SLOPE * a;
}

// order-preserving float<->uint encoding for atomicMax
__device__ __forceinline__ unsigned fenc(float f) {
    unsigned u = __float_as_uint(f);
    return (u & 0x80000000u) ? ~u : (u | 0x80000000u);
}
__device__ __forceinline__ float fdec(unsigned u) {
    u = (u & 0x80000000u) ? (u & 0x7fffffffu) : ~u;
    return __uint_as_float(u);
}

__device__ __forceinline__ void edge_sd(const int* ei, int e, int E, int& s, int& d) {
    if (e < E) { s = ei[e]; d = ei[E + e]; }
    else       { s = d = e - E; }          // appended self loops
}

// ---------------- kernels ----------------

__global__ void zero_ws_kernel(float* p, long long n) {
    long long t = (long long)blockIdx.x * blockDim.x + threadIdx.x;
    if (t < n) p[t] = 0.0f;
}

// Wt[n][k] = W1[k][n], K zero-padded to KPAD (removes all bounds checks from GEMM K loop)
__global__ void pad_w_kernel(const float* __restrict__ W, float* __restrict__ Wt) {
    int t = blockIdx.x * blockDim.x + threadIdx.x;
    if (t >= HC * KPAD) return;
    int n = t / KPAD, k = t % KPAD;
    Wt[t] = (k < F_IN) ? W[(long long)k * HC + n] : 0.0f;
}

// H1[N,256] = X[N,165] @ W1[165,256] using V_WMMA_F32_16X16X4_F32.
// Block: 512 threads = 16 waves; block handles 16 rows, wave w handles cols [16w,16w+16).
// Inner loop is branch-free: ds_load_b64 (A) + global_load_b64 (B, transposed-padded) + wmma.
__global__ __launch_bounds__(512)
void gemm1_wmma_kernel(const float* __restrict__ X, const float* __restrict__ Wt,
                       float* __restrict__ H1, int N) {
    __shared__ float As[16 * LDS_STRIDE];

    const int row0 = blockIdx.x * 16;
    const int tid  = threadIdx.x;

    // cooperative load of 16xK strip of X into LDS (zero-padded)
    for (int i = tid; i < 16 * LDS_STRIDE; i += 512) {
        int r = i / LDS_STRIDE, k = i % LDS_STRIDE;
        float v = 0.0f;
        int gr = row0 + r;
        if (gr < N && k < F_IN) v = X[(long long)gr * F_IN + k];
        As[i] = v;
    }
    __syncthreads();

    const int wave  = tid >> 5;
    const int lane  = tid & 31;
    const int m     = lane & 15;     // M (A frag) / N (B,C frags) within tile
    const int khalf = lane >> 4;     // 0: K={0,1}, 1: K={2,3}
    const int col0  = wave * 16;
    const int n     = col0 + m;

    const float* arow = &As[m * LDS_STRIDE];
    const float* brow = &Wt[(long long)n * KPAD];

    v8f acc = {};
    for (int k0 = 0; k0 < KPAD; k0 += 4) {
        const int kk = k0 + 2 * khalf;
        // A 16x4 f32 layout: lanes 0-15 hold (m, k0..k0+1), lanes 16-31 hold (m, k0+2..k0+3)
        v2f a = *(const v2f*)&arow[kk];
        // B 4x16 f32 layout: vgpr r: lanes0-15 K=k0+r, lanes16-31 K=k0+2+r; N=lane&15
        v2f b = *(const v2f*)&brow[kk];
        acc = __builtin_amdgcn_wmma_f32_16x16x4_f32(
            false, a, false, b, (short)0, acc, false, false);
    }

    // C/D 16x16 f32 layout: vgpr v: lanes0-15 M=v, lanes16-31 M=v+8; N=lane&15
    for (int v = 0; v < 8; ++v) {
        int r = row0 + v + 8 * khalf;
        if (r < N) H1[(long long)r * HC + n] = acc[v];
    }
}

// per-(node,head) attention scalars
__global__ void att1_kernel(const float* __restrict__ H1,
                            const float* __restrict__ att_s, const float* __restrict__ att_d,
                            float* __restrict__ a1s, float* __restrict__ a1d, int N) {
    int t = blockIdx.x * blockDim.x + threadIdx.x;
    if (t >= N * 4) return;
    int nd = t >> 2, h = t & 3;
    const float* hp = H1 + (long long)nd * HC + h * 64;
    const float* as = att_s + h * 64;
    const float* ad = att_d + h * 64;
    float s = 0.0f, d = 0.0f;
    for (int c = 0; c < 64; ++c) { float v = hp[c]; s = fmaf(v, as[c], s); d = fmaf(v, ad[c], d); }
    a1s[t] = s; a1d[t] = d;
}

__global__ void edge_max1_kernel(const int* __restrict__ ei, int E, int N,
                                 const float* __restrict__ a1s, const float* __restrict__ a1d,
                                 unsigned* __restrict__ segmax) {
    int t = blockIdx.x * blockDim.x + threadIdx.x;
    if (t >= (E + N) * 4) return;
    int e = t >> 2, h = t & 3, s, d;
    edge_sd(ei, e, E, s, d);
    float l = leaky(a1s[s * 4 + h] + a1d[d * 4 + h]);
    atomicMax(&segmax[d * 4 + h], fenc(l));
}

__global__ void edge_sum1_kernel(const int* __restrict__ ei, int E, int N,
                                 const float* __restrict__ a1s, const float* __restrict__ a1d,
                                 const unsigned* __restrict__ segmax, float* __restrict__ denom) {
    int t = blockIdx.x * blockDim.x + threadIdx.x;
    if (t >= (E + N) * 4) return;
    int e = t >> 2, h = t & 3, s, d;
    edge_sd(ei, e, E, s, d);
    float l = leaky(a1s[s * 4 + h] + a1d[d * 4 + h]);
    atomicAdd(&denom[d * 4 + h], __expf(l - fdec(segmax[d * 4 + h])));
}

// one thread per (edge, head, channel): coalesced gather + coalesced atomicAdd
__global__ void edge_agg1_kernel(const int* __restrict__ ei, int E, int N,
                                 const float* __restrict__ a1s, const float* __restrict__ a1d,
                                 const unsigned* __restrict__ segmax, const float* __restrict__ denom,
                                 const float* __restrict__ H1, float* __restrict__ O1) {
    long long t = (long long)blockIdx.x * blockDim.x + threadIdx.x;
    long long total = (long long)(E + N) * 256;
    if (t >= total) return;
    int c = (int)(t & 63);
    int h = (int)((t >> 6) & 3);
    int e = (int)(t >> 8);
    int s, d;
    edge_sd(ei, e, E, s, d);
    float l = leaky(a1s[s * 4 + h] + a1d[d * 4 + h]);
    float alpha = __expf(l - fdec(segmax[d * 4 + h])) / denom[d * 4 + h];
    atomicAdd(&O1[(long long)d * HC + h * 64 + c],
              H1[(long long)s * HC + h * 64 + c] * alpha);
}

// fused: relu(O1+b1) @ W2 -> G[N,2]; attention scalars for layer 2. One wave per node.
__global__ void node2_kernel(const float* __restrict__ O1, const float* __restrict__ b1,
                             const float* __restrict__ W2,
                             const float* __restrict__ as2, const float* __restrict__ ad2,
                             float* __restrict__ G, float* __restrict__ a2s, float* __restrict__ a2d,
                             int N) {
    int wid  = (blockIdx.x * blockDim.x + threadIdx.x) >> 5;
    int lane = threadIdx.x & 31;
    if (wid >= N) return;
    const float* row = O1 + (long long)wid * HC;
    float g0 = 0.0f, g1 = 0.0f;
    for (int c = lane; c < HC; c += 32) {
        float v = fmaxf(row[c] + b1[c], 0.0f);
        g0 = fmaf(v, W2[c * 2 + 0], g0);
        g1 = fmaf(v, W2[c * 2 + 1], g1);
    }
    for (int off = 16; off > 0; off >>= 1) {
        g0 += __shfl_xor(g0, off, 32);
        g1 += __shfl_xor(g1, off, 32);
    }
    if (lane == 0) {
        G[wid * 2 + 0] = g0;
        G[wid * 2 + 1] = g1;
        a2s[wid] = g0 * as2[0] + g1 * as2[1];
        a2d[wid] = g0 * ad2[0] + g1 * ad2[1];
    }
}

__global__ void edge_max2_kernel(const int* __restrict__ ei, int E, int N,
                                 const float* __restrict__ a2s, const float* __restrict__ a2d,
                                 unsigned* __restrict__ segmax) {
    int e = blockIdx.x * blockDim.x + threadIdx.x;
    if (e >= E + N) return;
    int s, d; edge_sd(ei, e, E, s, d);
    atomicMax(&segmax[d], fenc(leaky(a2s[s] + a2d[d])));
}

__global__ void edge_sum2_kernel(const int* __restrict__ ei, int E, int N,
                                 const float* __restrict__ a2s, const float* __restrict__ a2d,
                                 const unsigned* __restrict__ segmax, float* __restrict__ denom) {
    int e = blockIdx.x * blockDim.x + threadIdx.x;
    if (e >= E + N) return;
    int s, d; edge_sd(ei, e, E, s, d);
    float l = leaky(a2s[s] + a2d[d]);
    atomicAdd(&denom[d], __expf(l - fdec(segmax[d])));
}

__global__ void edge_agg2_kernel(const int* __restrict__ ei, int E, int N,
                                 const float* __restrict__ a2s, const float* __restrict__ a2d,
                                 const unsigned* __restrict__ segmax, const float* __restrict__ denom,
                                 const float* __restrict__ G, float* __restrict__ O2) {
    int e = blockIdx.x * blockDim.x + threadIdx.x;
    if (e >= E + N) return;
    int s, d; edge_sd(ei, e, E, s, d);
    float l = leaky(a2s[s] + a2d[d]);
    float alpha = __expf(l - fdec(segmax[d])) / denom[d];
    atomicAdd(&O2[d * 2 + 0], G[s * 2 + 0] * alpha);
    atomicAdd(&O2[d * 2 + 1], G[s * 2 + 1] * alpha);
}

__global__ void finalize_kernel(const float* __restrict__ O2, const float* __restrict__ b2,
                                float* __restrict__ out, int N) {
    int nd = blockIdx.x * blockDim.x + threadIdx.x;
    if (nd >= N) return;
    float v0 = O2[nd * 2 + 0] + b2[0];
    float v1 = O2[nd * 2 + 1] + b2[1];
    float m  = fmaxf(v0, v1);
    float lse = m + logf(__expf(v0 - m) + __expf(v1 - m));
    out[nd * 2 + 0] = v0 - lse;
    out[nd * 2 + 1] = v1 - lse;
}

// ---------------- launch ----------------

extern "C" void kernel_launch(void* const* d_in, const int* in_sizes, int n_in,
                              void* d_out, int out_size, void* d_ws, size_t ws_size,
                              hipStream_t stream) {
    const float* x   = (const float*)d_in[0];
    const int*   ei  = (const int*)  d_in[1];
    const float* W1  = (const float*)d_in[2];
    const float* as1 = (const float*)d_in[3];
    const float* ad1 = (const float*)d_in[4];
    const float* b1  = (const float*)d_in[5];
    const float* W2  = (const float*)d_in[6];
    const float* as2 = (const float*)d_in[7];
    const float* ad2 = (const float*)d_in[8];
    const float* b2  = (const float*)d_in[9];
    float* out = (float*)d_out;

    const int N = in_sizes[0] / F_IN;
    const int E = in_sizes[1] / 2;
    const int EN = E + N;

    // workspace layout: [zeroed region | persistent region]
    float* ws = (float*)d_ws;
    long long off = 0;
    float*    O1      = ws + off;              off += (long long)N * HC;
    unsigned* segmax1 = (unsigned*)(ws + off); off += (long long)N * 4;
    float*    denom1  = ws + off;              off += (long long)N * 4;
    float*    O2      = ws + off;              off += (long long)N * 2;
    unsigned* segmax2 = (unsigned*)(ws + off); off += N;
    float*    denom2  = ws + off;              off += N;
    const long long zero_elems = off;          // everything above starts at 0 each call
    float* H1  = ws + off;                     off += (long long)N * HC;
    float* a1s = ws + off;                     off += (long long)N * 4;
    float* a1d = ws + off;                     off += (long long)N * 4;
    float* G   = ws + off;                     off += (long long)N * 2;
    float* a2s = ws + off;                     off += N;
    float* a2d = ws + off;                     off += N;
    float* Wt  = ws + off;                     off += (long long)HC * KPAD;

    const int B = 256;

    zero_ws_kernel<<<dim3((unsigned)((zero_elems + B - 1) / B)), B, 0, stream>>>(ws, zero_elems);

    pad_w_kernel<<<dim3((HC * KPAD + B - 1) / B), B, 0, stream>>>(W1, Wt);

    gemm1_wmma_kernel<<<dim3((N + 15) / 16), 512, 0, stream>>>(x, Wt, H1, N);

    att1_kernel<<<dim3((N * 4 + B - 1) / B), B, 0, stream>>>(H1, as1, ad1, a1s, a1d, N);

    edge_max1_kernel<<<dim3((EN * 4 + B - 1) / B), B, 0, stream>>>(ei, E, N, a1s, a1d, segmax1);
    edge_sum1_kernel<<<dim3((EN * 4 + B - 1) / B), B, 0, stream>>>(ei, E, N, a1s, a1d, segmax1, denom1);
    {
        long long total = (long long)EN * 256;
        edge_agg1_kernel<<<dim3((unsigned)((total + B - 1) / B)), B, 0, stream>>>(
            ei, E, N, a1s, a1d, segmax1, denom1, H1, O1);
    }

    node2_kernel<<<dim3((N * 32 + B - 1) / B), B, 0, stream>>>(O1, b1, W2, as2, ad2, G, a2s, a2d, N);

    edge_max2_kernel<<<dim3((EN + B - 1) / B), B, 0, stream>>>(ei, E, N, a2s, a2d, segmax2);
    edge_sum2_kernel<<<dim3((EN + B - 1) / B), B, 0, stream>>>(ei, E, N, a2s, a2d, segmax2, denom2);
    edge_agg2_kernel<<<dim3((EN + B - 1) / B), B, 0, stream>>>(ei, E, N, a2s, a2d, segmax2, denom2, G, O2);

    finalize_kernel<<<dim3((N + B - 1) / B), B, 0, stream>>>(O2, b2, out, N);
}